// RNN_1168231104732
// MI455X (gfx1250) — compile-verified
//
#include <hip/hip_runtime.h>
#include <math.h>

typedef __attribute__((ext_vector_type(2))) float v2f;
typedef __attribute__((ext_vector_type(8))) float v8f;

#define DDIM 64   // input feature dim (K)
#define HDIM 2    // hidden dim (N)
#define TILE_M 16 // WMMA M tile

// ---------------------------------------------------------------------------
// Phase 1: Z[r, n] = X[r, :] . W_ih[n, :] + b_ih[n] + b_hh[n],  r in [0, T*B)
// One wave per 16-row tile; 16 chained V_WMMA_F32_16X16X4_F32 (K=4 each).
// Exact fp32 — matches the fp32 reference path.
// ---------------------------------------------------------------------------
__global__ void rnn_gemm_wmma(const float* __restrict__ X,    // [R, 64]
                              const float* __restrict__ W,    // [2, 64]
                              const float* __restrict__ b_ih, // [2]
                              const float* __restrict__ b_hh, // [2]
                              float* __restrict__ Z,          // [R, 2]
                              long R)
{
    const int lane = threadIdx.x & 31;
    const int wave = threadIdx.x >> 5;
    const long tile = (long)blockIdx.x * (blockDim.x >> 5) + wave;
    const long row0 = tile * TILE_M;
    if (row0 >= R) return; // uniform per-wave

    const int nl = lane & 15;         // A: row-in-tile (M); B/D: column (N)
    const int kh = (lane >> 4) << 1;  // K sub-offset: 0 (lanes 0-15) or 2 (16-31)

    // A operand: lane reads its row's floats at columns {kh+k, kh+k+1}, k = 0,4,8,...
    const v2f* __restrict__ arow =
        reinterpret_cast<const v2f*>(X + (row0 + nl) * DDIM + kh);

    // B operand: column N=nl of W_ih^T padded to 16 columns (zeros for N>=2).
    const float wmask = (nl < HDIM) ? 1.0f : 0.0f;
    const float* __restrict__ wrow = W + (nl < HDIM ? nl : 0) * DDIM + kh;

    v8f acc = {0.f, 0.f, 0.f, 0.f, 0.f, 0.f, 0.f, 0.f};
#pragma unroll
    for (int k = 0; k < DDIM; k += 4) {
        v2f a = arow[k >> 1];          // floats {kh+k, kh+k+1} of this row
        v2f bm;
        bm.x = wrow[k + 0] * wmask;
        bm.y = wrow[k + 1] * wmask;
        // D = A(16x4 f32) x B(4x16 f32) + C(16x16 f32)
        acc = __builtin_amdgcn_wmma_f32_16x16x4_f32(
            false, a, false, bm, (short)0, acc, false, false);
    }

    // D layout: lane L holds column N = L%16, rows M = 8*(L/16) + r (r = VGPR idx).
    if (nl < HDIM) {
        const float bias = b_ih[nl] + b_hh[nl];
        const long mbase = row0 + ((lane >> 4) << 3);
#pragma unroll
        for (int r = 0; r < 8; ++r) {
            Z[(mbase + r) * HDIM + nl] = acc[r] + bias;
        }
    }
}

// ---------------------------------------------------------------------------
// Phase 2: sequential tanh recurrence. One lane per batch element; 8 single-
// wave blocks so Z reads are fetched by 8 WGPs in parallel. Loads of z_t are
// independent of h -> deep prefetch hides L2/HBM latency behind the
// FMA->FMA->tanh dependency chain.
// ---------------------------------------------------------------------------
__global__ void rnn_scan(const float* __restrict__ Z,    // [T*B, 2]
                         const float* __restrict__ W_hh, // [2, 2]
                         float* __restrict__ out,        // [B, 2]
                         int T, int B)
{
    const int b = blockIdx.x * blockDim.x + threadIdx.x;
    if (b >= B) return;

    const float w00 = W_hh[0], w01 = W_hh[1]; // h_new[0] = tanh(z0 + h0*w00 + h1*w01)
    const float w10 = W_hh[2], w11 = W_hh[3]; // h_new[1] = tanh(z1 + h0*w10 + h1*w11)

    const float2* __restrict__ zp = reinterpret_cast<const float2*>(Z) + b;

    float h0 = 0.f, h1 = 0.f;
    for (int t = 0; t < T; ++t) {
        if (t + 24 < T)
            __builtin_prefetch(zp + (size_t)(t + 24) * B, 0, 0); // global_prefetch_b8
        float2 z = zp[(size_t)t * B]; // coalesced: consecutive lanes, consecutive 8B
        float a0 = z.x + h0 * w00 + h1 * w01;
        float a1 = z.y + h0 * w10 + h1 * w11;
        h0 = tanhf(a0);
        h1 = tanhf(a1);
    }
    out[b * HDIM + 0] = h0;
    out[b * HDIM + 1] = h1;
}

// ---------------------------------------------------------------------------
extern "C" void kernel_launch(void* const* d_in, const int* in_sizes, int n_in,
                              void* d_out, int out_size, void* d_ws, size_t ws_size,
                              hipStream_t stream) {
    (void)n_in; (void)ws_size;
    const float* X    = (const float*)d_in[0]; // [T, B, D] fp32
    const float* W_ih = (const float*)d_in[1]; // [H, D]
    const float* W_hh = (const float*)d_in[2]; // [H, H]
    const float* b_ih = (const float*)d_in[3]; // [H]
    const float* b_hh = (const float*)d_in[4]; // [H]
    float* out = (float*)d_out;                // [B, H]

    const long R = (long)in_sizes[0] / DDIM;   // T*B = 1,048,576
    const int  B = out_size / HDIM;            // 256
    const int  T = (int)(R / B);               // 4096

    float* Z = (float*)d_ws;                   // R*HDIM fp32 = 8 MB scratch

    // Phase 1: 65536 tiles, 8 waves (256 threads) per block.
    const int waves_per_block = 8;
    const long tiles = R / TILE_M;
    dim3 g1((unsigned)((tiles + waves_per_block - 1) / waves_per_block));
    rnn_gemm_wmma<<<g1, waves_per_block * 32, 0, stream>>>(X, W_ih, b_ih, b_hh, Z, R);

    // Phase 2: one lane per batch element, 1 wave per block across 8 WGPs.
    dim3 g2((unsigned)((B + 31) / 32));
    rnn_scan<<<g2, 32, 0, stream>>>(Z, W_hh, out, T, B);
}